// GravNetBlock_4501125726945
// MI455X (gfx1250) — compile-verified
//
#include <hip/hip_runtime.h>
#include <math.h>

// ---------------------------------------------------------------------------
// GravNet block for MI455X (gfx1250, wave32).
// f32 WMMA (V_WMMA_F32_16X16X4_F32) for all dense GEMMs; LDS-resident learned
// coords for the O(n^2) kNN; register-resident top-K=40 lists; L2-resident
// gather of propagate features.
// ---------------------------------------------------------------------------

typedef __attribute__((ext_vector_type(2))) float v2f;
typedef __attribute__((ext_vector_type(8))) float v8f;

#define NB    32            // graphs
#define NPG   2048          // nodes per graph
#define NTOT  (NB * NPG)    // 65536
#define IN_C  64
#define DSF   32            // d_shape
#define SD    3             // space dims
#define PD    64            // propagate dims
#define KNN   40

__device__ __forceinline__ float elu(float v) {
    return v > 0.f ? v : (expf(v) - 1.f);
}

__device__ __forceinline__ v8f wmma4(v8f c, v2f a, v2f b) {
    // D = A(16x4 f32) * B(4x16 f32) + C(16x16 f32), wave32
    return __builtin_amdgcn_wmma_f32_16x16x4_f32(
        /*neg_a=*/false, a, /*neg_b=*/false, b,
        /*c_mod=*/(short)0, c, /*reuse_a=*/false, /*reuse_b=*/false);
}

// A-matrix 16x4 f32 fragment from a row-major [16 x lda] tile.
// lanes 0-15: M=lane, K = k..k+1 ; lanes 16-31: M=lane-16, K = k+2..k+3
__device__ __forceinline__ v2f frag_a(const float* A, int lda, int k, int lane) {
    int m  = lane & 15;
    int ko = k + ((lane >> 4) << 1);
    v2f a;
    a.x = A[m * lda + ko];
    a.y = A[m * lda + ko + 1];
    return a;
}

// B-matrix 4x16 f32 fragment from row-major weights W[k][n] with leading dim ldb.
__device__ __forceinline__ v2f frag_b(const float* W, int ldb, int k, int n0, int lane) {
    int n  = n0 + (lane & 15);
    int ko = k + ((lane >> 4) << 1);
    v2f b;
    b.x = W[ko * ldb + n];
    b.y = W[(ko + 1) * ldb + n];
    return b;
}

// ---------------------------------------------------------------------------
// Kernel 1: pre-MLP (64->32 ELU ->32 ELU), s = x1@s_w+b (3D + |s|^2 packed),
// h = x1@h_w+b (64D).  128 threads = 4 waves, 16 nodes per wave.
// ---------------------------------------------------------------------------
__global__ __launch_bounds__(128) void k_pre(
    const float* __restrict__ x,
    const float* __restrict__ w1, const float* __restrict__ b1,
    const float* __restrict__ w2, const float* __restrict__ b2,
    const float* __restrict__ sw, const float* __restrict__ sb,
    const float* __restrict__ hw, const float* __restrict__ hb,
    float* __restrict__ x1o, float4* __restrict__ s4o, float* __restrict__ ho)
{
    __shared__ float w1s[IN_C * DSF];
    __shared__ float w2s[DSF * DSF];
    __shared__ float hws[DSF * PD];
    __shared__ float sws[DSF * SD];
    __shared__ float b1s[DSF], b2s[DSF], hbs[PD], sbs[SD];
    __shared__ float xt [4][16 * IN_C];
    __shared__ float t1 [4][16 * DSF];
    __shared__ float x1t[4][16 * DSF];

    const int tid  = threadIdx.x;
    const int w    = tid >> 5;
    const int lane = tid & 31;
    const int nbase = blockIdx.x * 64 + w * 16;

    for (int i = tid; i < IN_C * DSF; i += 128) w1s[i] = w1[i];
    for (int i = tid; i < DSF * DSF;  i += 128) w2s[i] = w2[i];
    for (int i = tid; i < DSF * PD;   i += 128) hws[i] = hw[i];
    for (int i = tid; i < DSF * SD;   i += 128) sws[i] = sw[i];
    if (tid < DSF) { b1s[tid] = b1[tid]; b2s[tid] = b2[tid]; }
    if (tid < PD)  hbs[tid] = hb[tid];
    if (tid < SD)  sbs[tid] = sb[tid];

    const float* xg = x + (size_t)nbase * IN_C;
    for (int i = lane; i < 16 * IN_C; i += 32) xt[w][i] = xg[i];
    __syncthreads();

    float* Xw = xt[w];
    float* T1 = t1[w];
    float* X1 = x1t[w];

    // GEMM1: [16x64] @ [64x32] + b1, ELU
    for (int n0 = 0; n0 < DSF; n0 += 16) {
        v8f c = {};
        for (int k = 0; k < IN_C; k += 4)
            c = wmma4(c, frag_a(Xw, IN_C, k, lane), frag_b(w1s, DSF, k, n0, lane));
        int n = n0 + (lane & 15);
        #pragma unroll
        for (int i = 0; i < 8; ++i) {
            int m = i + ((lane >> 4) << 3);
            T1[m * DSF + n] = elu(c[i] + b1s[n]);
        }
    }
    __syncthreads();

    // GEMM2: [16x32] @ [32x32] + b2, ELU  -> x1
    for (int n0 = 0; n0 < DSF; n0 += 16) {
        v8f c = {};
        for (int k = 0; k < DSF; k += 4)
            c = wmma4(c, frag_a(T1, DSF, k, lane), frag_b(w2s, DSF, k, n0, lane));
        int n = n0 + (lane & 15);
        #pragma unroll
        for (int i = 0; i < 8; ++i) {
            int m = i + ((lane >> 4) << 3);
            float v = elu(c[i] + b2s[n]);
            X1[m * DSF + n] = v;
            x1o[(size_t)(nbase + m) * DSF + n] = v;
        }
    }
    __syncthreads();

    // h = x1 @ h_w[32x64] + hb
    for (int n0 = 0; n0 < PD; n0 += 16) {
        v8f c = {};
        for (int k = 0; k < DSF; k += 4)
            c = wmma4(c, frag_a(X1, DSF, k, lane), frag_b(hws, PD, k, n0, lane));
        int n = n0 + (lane & 15);
        #pragma unroll
        for (int i = 0; i < 8; ++i) {
            int m = i + ((lane >> 4) << 3);
            ho[(size_t)(nbase + m) * PD + n] = c[i] + hbs[n];
        }
    }

    // s = x1 @ s_w[32x3] + sb ; pack (s0,s1,s2,|s|^2)
    if (lane < 16) {
        float s0 = sbs[0], s1 = sbs[1], s2 = sbs[2];
        for (int k = 0; k < DSF; ++k) {
            float xv = X1[lane * DSF + k];
            s0 += xv * sws[k * SD + 0];
            s1 += xv * sws[k * SD + 1];
            s2 += xv * sws[k * SD + 2];
        }
        float sq = s0 * s0 + s1 * s1 + s2 * s2;
        s4o[nbase + lane] = make_float4(s0, s1, s2, sq);
    }
}

// ---------------------------------------------------------------------------
// Kernel 2: per-graph kNN (K=40) over 2048 nodes in learned 3-space.
// Coordinates LDS-resident; top-K list register-resident with fully-unrolled
// compare/swap insertion.  256 blocks (8 per graph), 256 threads, 1 node each.
// ---------------------------------------------------------------------------
__global__ __launch_bounds__(256) void k_knn(
    const float4* __restrict__ s4,
    int* __restrict__ graph_out,       // [NB][NPG][KNN] local indices
    float* __restrict__ wbuf)          // [NTOT][KNN] edge weights
{
    __shared__ float4 st[NPG];

    const int b8 = blockIdx.x;
    const int g  = b8 >> 3;
    const int nl = (b8 & 7) * 256 + threadIdx.x;   // node local to graph

    for (int i = threadIdx.x; i < NPG; i += 256)
        st[i] = s4[(size_t)g * NPG + i];
    __syncthreads();

    float kd[KNN];
    int   ki[KNN];
    #pragma unroll
    for (int t = 0; t < KNN; ++t) { kd[t] = 3.4e38f; ki[t] = 0; }

    const float4 me = st[nl];
    for (int j = 0; j < NPG; ++j) {
        float4 o = st[j];
        float d2 = me.w + o.w - 2.f * (me.x * o.x + me.y * o.y + me.z * o.z);
        if (d2 < kd[KNN - 1]) {
            kd[KNN - 1] = d2;
            ki[KNN - 1] = j;
            #pragma unroll
            for (int t = KNN - 1; t > 0; --t) {
                if (kd[t] < kd[t - 1]) {
                    float td = kd[t]; kd[t] = kd[t - 1]; kd[t - 1] = td;
                    int   ti = ki[t]; ki[t] = ki[t - 1]; ki[t - 1] = ti;
                }
            }
        }
    }

    int*   gp = graph_out + ((size_t)g * NPG + nl) * KNN;
    float* wp = wbuf      + ((size_t)g * NPG + nl) * KNN;
    #pragma unroll
    for (int t = 0; t < KNN; ++t) {
        gp[t] = ki[t];
        wp[t] = expf(-10.f * fmaxf(kd[t], 0.f));
    }
}

// ---------------------------------------------------------------------------
// Kernel 3: gather neighbors, weighted mean/max aggregation, assemble
// feat = [x1 | mean | max]  ([NTOT][160]).  One wave per node; lane = 2 chans.
// ---------------------------------------------------------------------------
__global__ __launch_bounds__(256) void k_gather(
    const int* __restrict__ graph_out,
    const float* __restrict__ wbuf,
    const float* __restrict__ h,
    const float* __restrict__ x1,
    float* __restrict__ feat)
{
    const int tid  = threadIdx.x;
    const int w    = tid >> 5;
    const int lane = tid & 31;
    const int node = blockIdx.x * 8 + w;
    const int g    = node / NPG;

    const int*   gp = graph_out + (size_t)node * KNN;
    const float* wp = wbuf + (size_t)node * KNN;
    const float* hb = h + (size_t)g * NPG * PD;

    float m0 = 0.f, m1 = 0.f;
    float xx0 = -3.4e38f, xx1 = -3.4e38f;
    for (int k = 0; k < KNN; ++k) {
        int   nb = gp[k];
        float wk = wp[k];
        const float* hp = hb + (size_t)nb * PD + 2 * lane;
        float v0 = hp[0] * wk;
        float v1 = hp[1] * wk;
        m0 += v0; m1 += v1;
        xx0 = fmaxf(xx0, v0); xx1 = fmaxf(xx1, v1);
    }
    m0 *= (1.f / KNN);
    m1 *= (1.f / KNN);

    float* f = feat + (size_t)node * 160;
    f[DSF + 2 * lane]          = m0;
    f[DSF + 2 * lane + 1]      = m1;
    f[DSF + PD + 2 * lane]     = xx0;
    f[DSF + PD + 2 * lane + 1] = xx1;
    if (lane < 16) {
        f[2 * lane]     = x1[(size_t)node * DSF + 2 * lane];
        f[2 * lane + 1] = x1[(size_t)node * DSF + 2 * lane + 1];
    }
}

// ---------------------------------------------------------------------------
// Kernel 4: gout = feat @ out_w[160x32] + b ; concat s -> [36] (zero padded);
// post MLP (36->32 ELU -> 32 ELU) -> p.   128 threads, 16 nodes / wave, WMMA.
// ---------------------------------------------------------------------------
__global__ __launch_bounds__(128) void k_mlp(
    const float* __restrict__ feat,
    const float4* __restrict__ s4,
    const float* __restrict__ ow, const float* __restrict__ ob,
    const float* __restrict__ pw1, const float* __restrict__ pb1,
    const float* __restrict__ pw2, const float* __restrict__ pb2,
    float* __restrict__ p)
{
    __shared__ float ows[160 * DSF];
    __shared__ float pw1s[36 * DSF];
    __shared__ float pw2s[DSF * DSF];
    __shared__ float obs[DSF], pb1s[DSF], pb2s[DSF];
    __shared__ float pin[4][16 * 36];
    __shared__ float tt [4][16 * DSF];

    const int tid  = threadIdx.x;
    const int w    = tid >> 5;
    const int lane = tid & 31;
    const int nbase = blockIdx.x * 64 + w * 16;

    for (int i = tid; i < 160 * DSF; i += 128) ows[i] = ow[i];
    for (int i = tid; i < 36 * DSF;  i += 128) pw1s[i] = (i < 35 * DSF) ? pw1[i] : 0.f;
    for (int i = tid; i < DSF * DSF; i += 128) pw2s[i] = pw2[i];
    if (tid < DSF) { obs[tid] = ob[tid]; pb1s[tid] = pb1[tid]; pb2s[tid] = pb2[tid]; }
    __syncthreads();

    float* PIN = pin[w];
    float* TT  = tt[w];
    const float* Af = feat + (size_t)nbase * 160;

    // gout = feat[16x160] @ out_w + ob
    for (int n0 = 0; n0 < DSF; n0 += 16) {
        v8f c = {};
        for (int k = 0; k < 160; k += 4)
            c = wmma4(c, frag_a(Af, 160, k, lane), frag_b(ows, DSF, k, n0, lane));
        int n = n0 + (lane & 15);
        #pragma unroll
        for (int i = 0; i < 8; ++i) {
            int m = i + ((lane >> 4) << 3);
            PIN[m * 36 + n] = c[i] + obs[n];
        }
    }
    // append s (3) + zero pad
    if (lane < 16) {
        float4 sv = s4[nbase + lane];
        PIN[lane * 36 + 32] = sv.x;
        PIN[lane * 36 + 33] = sv.y;
        PIN[lane * 36 + 34] = sv.z;
        PIN[lane * 36 + 35] = 0.f;
    }
    __syncthreads();

    // post MLP layer 1: [16x36] @ [36x32] + b, ELU
    for (int n0 = 0; n0 < DSF; n0 += 16) {
        v8f c = {};
        for (int k = 0; k < 36; k += 4)
            c = wmma4(c, frag_a(PIN, 36, k, lane), frag_b(pw1s, DSF, k, n0, lane));
        int n = n0 + (lane & 15);
        #pragma unroll
        for (int i = 0; i < 8; ++i) {
            int m = i + ((lane >> 4) << 3);
            TT[m * DSF + n] = elu(c[i] + pb1s[n]);
        }
    }
    __syncthreads();

    // post MLP layer 2: [16x32] @ [32x32] + b, ELU -> p
    for (int n0 = 0; n0 < DSF; n0 += 16) {
        v8f c = {};
        for (int k = 0; k < DSF; k += 4)
            c = wmma4(c, frag_a(TT, DSF, k, lane), frag_b(pw2s, DSF, k, n0, lane));
        int n = n0 + (lane & 15);
        #pragma unroll
        for (int i = 0; i < 8; ++i) {
            int m = i + ((lane >> 4) << 3);
            p[(size_t)(nbase + m) * DSF + n] = elu(c[i] + pb2s[n]);
        }
    }
}

// ---------------------------------------------------------------------------
// Kernel 5: per-graph mean/min/max of p over the 2048 nodes -> gstats[b][3][32]
// ---------------------------------------------------------------------------
__global__ __launch_bounds__(256) void k_stats(
    const float* __restrict__ p, float* __restrict__ gstats)
{
    __shared__ float rs[8][DSF], rn[8][DSF], rx[8][DSF];
    const int b   = blockIdx.x;
    const int c   = threadIdx.x & 31;
    const int grp = threadIdx.x >> 5;

    const float* pg = p + (size_t)b * NPG * DSF;
    float sum = 0.f, mn = 3.4e38f, mx = -3.4e38f;
    for (int r = grp; r < NPG; r += 8) {
        float v = pg[(size_t)r * DSF + c];
        sum += v;
        mn = fminf(mn, v);
        mx = fmaxf(mx, v);
    }
    rs[grp][c] = sum; rn[grp][c] = mn; rx[grp][c] = mx;
    __syncthreads();

    if (threadIdx.x < DSF) {
        float s = 0.f, n = 3.4e38f, x = -3.4e38f;
        #pragma unroll
        for (int gi = 0; gi < 8; ++gi) {
            s += rs[gi][c];
            n = fminf(n, rn[gi][c]);
            x = fmaxf(x, rx[gi][c]);
        }
        gstats[b * 96 + c]      = s * (1.f / NPG);
        gstats[b * 96 + 32 + c] = n;
        gstats[b * 96 + 64 + c] = x;
    }
}

// ---------------------------------------------------------------------------
// Kernel 6: out = ELU( [gmean|gmin|gmax|p] (128) @ fin_w[128x32] + fb )
// A fragment is synthesized per lane: k<96 from graph stats, k>=96 from p.
// ---------------------------------------------------------------------------
__global__ __launch_bounds__(128) void k_final(
    const float* __restrict__ p,
    const float* __restrict__ gstats,
    const float* __restrict__ fw, const float* __restrict__ fb,
    float* __restrict__ out)
{
    __shared__ float fws[128 * DSF];
    __shared__ float fbs[DSF];
    __shared__ float gss[96];

    const int tid  = threadIdx.x;
    const int w    = tid >> 5;
    const int lane = tid & 31;
    const int nbase = blockIdx.x * 64 + w * 16;
    const int g     = nbase / NPG;

    for (int i = tid; i < 128 * DSF; i += 128) fws[i] = fw[i];
    if (tid < DSF) fbs[tid] = fb[tid];
    if (tid < 96)  gss[tid] = gstats[g * 96 + tid];
    __syncthreads();

    const int m = lane & 15;
    const float* prow = p + (size_t)(nbase + m) * DSF;

    for (int n0 = 0; n0 < DSF; n0 += 16) {
        v8f c = {};
        for (int k = 0; k < 128; k += 4) {
            int ko = k + ((lane >> 4) << 1);
            v2f a;
            a.x = (ko < 96)     ? gss[ko]         : prow[ko - 96];
            a.y = (ko + 1 < 96) ? gss[ko + 1]     : prow[ko + 1 - 96];
            c = wmma4(c, a, frag_b(fws, DSF, k, n0, lane));
        }
        int n = n0 + (lane & 15);
        #pragma unroll
        for (int i = 0; i < 8; ++i) {
            int mm = i + ((lane >> 4) << 3);
            out[(size_t)(nbase + mm) * DSF + n] = elu(c[i] + fbs[n]);
        }
    }
}

// ---------------------------------------------------------------------------
extern "C" void kernel_launch(void* const* d_in, const int* in_sizes, int n_in,
                              void* d_out, int out_size, void* d_ws, size_t ws_size,
                              hipStream_t stream) {
    const float* x   = (const float*)d_in[0];
    // d_in[1] = batch (unused: layout is implicit, NPG nodes per graph)
    const float* w1  = (const float*)d_in[2];
    const float* b1  = (const float*)d_in[3];
    const float* w2  = (const float*)d_in[4];
    const float* b2  = (const float*)d_in[5];
    const float* sw  = (const float*)d_in[6];
    const float* sb  = (const float*)d_in[7];
    const float* hw  = (const float*)d_in[8];
    const float* hb  = (const float*)d_in[9];
    const float* ow  = (const float*)d_in[10];
    const float* ob  = (const float*)d_in[11];
    const float* pw1 = (const float*)d_in[12];
    const float* pb1 = (const float*)d_in[13];
    const float* pw2 = (const float*)d_in[14];
    const float* pb2 = (const float*)d_in[15];
    const float* fw  = (const float*)d_in[16];
    const float* fbv = (const float*)d_in[17];

    float* outF = (float*)d_out;
    int* graph_out = (int*)(outF + (size_t)NTOT * DSF);

    // workspace layout (floats)
    float* ws   = (float*)d_ws;
    float*  x1  = ws;                                   // NTOT*32
    float4* s4  = (float4*)(ws + (size_t)NTOT * DSF);   // NTOT*4 (16B aligned)
    float*  h   = ws + (size_t)NTOT * DSF + (size_t)NTOT * 4;   // NTOT*64
    float*  wb  = h + (size_t)NTOT * PD;                // NTOT*40
    float*  ft  = wb + (size_t)NTOT * KNN;              // NTOT*160
    float*  p   = ft + (size_t)NTOT * 160;              // NTOT*32
    float*  gst = p + (size_t)NTOT * DSF;               // NB*96

    k_pre   <<<NTOT / 64, 128, 0, stream>>>(x, w1, b1, w2, b2, sw, sb, hw, hb, x1, s4, h);
    k_knn   <<<NB * 8,    256, 0, stream>>>(s4, graph_out, wb);
    k_gather<<<NTOT / 8,  256, 0, stream>>>(graph_out, wb, h, x1, ft);
    k_mlp   <<<NTOT / 64, 128, 0, stream>>>(ft, s4, ow, ob, pw1, pb1, pw2, pb2, p);
    k_stats <<<NB,        256, 0, stream>>>(p, gst);
    k_final <<<NTOT / 64, 128, 0, stream>>>(p, gst, fw, fbv, outF);

    // trailing two scalar zeros of the output tuple
    hipMemsetAsync(outF + (size_t)out_size - 2, 0, 2 * sizeof(float), stream);
}